// GCN_2layer_11510512353338
// MI455X (gfx1250) — compile-verified
//
#include <hip/hip_runtime.h>
#include <hip/hip_bf16.h>

typedef float v2f __attribute__((ext_vector_type(2)));
typedef float v8f __attribute__((ext_vector_type(8)));

#define N_NODES   100000
#define N_EDGES   1600000
#define CHN       128
#define OUT_C     10
#define NUM_G     64

// Non-returning, relaxed, agent-scope f32 atomic add -> GLOBAL_ATOMIC_ADD_F32
__device__ __forceinline__ void atom_add_f32(float* p, float v) {
    (void)__hip_atomic_fetch_add(p, v, __ATOMIC_RELAXED, __HIP_MEMORY_SCOPE_AGENT);
}

// ---------------------------------------------------------------------------
// Degree count: one thread per edge, u32 atomic on dst
// ---------------------------------------------------------------------------
__global__ void gcn_deg_kernel(const int* __restrict__ dst, unsigned* __restrict__ deg, int E) {
    int e = blockIdx.x * blockDim.x + threadIdx.x;
    if (e < E) atomicAdd(&deg[dst[e]], 1u);
}

// dinv = rsqrt(deg + 1)   (+1 = self loop; always > 0)
__global__ void gcn_dinv_kernel(const unsigned* __restrict__ deg, float* __restrict__ dinv, int n) {
    int i = blockIdx.x * blockDim.x + threadIdx.x;
    if (i < n) dinv[i] = rsqrtf((float)(deg[i] + 1u));
}

// ---------------------------------------------------------------------------
// WMMA f32 GEMM:  Y[M x 128] = X[M x 128] @ W[128 x 128]
// One wave computes a 16-row strip across all 128 output columns.
// A layout (16x4 f32): lanes 0-15 -> K={ks,ks+1}, lanes 16-31 -> K={ks+2,ks+3}
// B layout (4x16 f32): V0/V1 hold rows K=kb,kb+1; col = lane&15; kb = ks+2*(lane>>4)
// C/D layout: VGPR v, lane L -> row v + 8*(L>>4), col L&15
// ---------------------------------------------------------------------------
__global__ void gcn_gemm128_wmma_kernel(const float* __restrict__ X,
                                        const float* __restrict__ W,
                                        float* __restrict__ Y, int M) {
    const int wave = threadIdx.x >> 5;
    const int lane = threadIdx.x & 31;
    const int m0 = (blockIdx.x * 8 + wave) * 16;
    if (m0 >= M) return;                      // M % 16 == 0, whole strip valid

    const int half = lane >> 4;               // 0 or 1
    const int r    = lane & 15;

    const float* arow = X + (m0 + r) * CHN + 2 * half;

    v8f acc0 = {}; v8f acc1 = {}; v8f acc2 = {}; v8f acc3 = {};
    v8f acc4 = {}; v8f acc5 = {}; v8f acc6 = {}; v8f acc7 = {};

    for (int ks = 0; ks < CHN; ks += 4) {
        v2f a;
        a.x = arow[ks];
        a.y = arow[ks + 1];
        const int kb = ks + 2 * half;
        const float* w0 = W + kb * CHN + r;       // row kb
        const float* w1 = w0 + CHN;               // row kb+1
#define GCN_WMMA_STEP(ACC, NT)                                                  \
        {                                                                       \
            v2f b;                                                              \
            b.x = w0[(NT) * 16];                                                \
            b.y = w1[(NT) * 16];                                                \
            ACC = __builtin_amdgcn_wmma_f32_16x16x4_f32(                        \
                false, a, false, b, (short)0, ACC, false, false);               \
        }
        GCN_WMMA_STEP(acc0, 0) GCN_WMMA_STEP(acc1, 1)
        GCN_WMMA_STEP(acc2, 2) GCN_WMMA_STEP(acc3, 3)
        GCN_WMMA_STEP(acc4, 4) GCN_WMMA_STEP(acc5, 5)
        GCN_WMMA_STEP(acc6, 6) GCN_WMMA_STEP(acc7, 7)
#undef GCN_WMMA_STEP
    }

    // store D: row = m0 + v + 8*half, col = nt*16 + r
    float* ybase = Y + (m0 + 8 * half) * CHN + r;
#define GCN_STORE_TILE(ACC, NT)                                                 \
        {                                                                       \
            _Pragma("unroll")                                                   \
            for (int v = 0; v < 8; ++v)                                         \
                ybase[v * CHN + (NT) * 16] = ACC[v];                            \
        }
    GCN_STORE_TILE(acc0, 0) GCN_STORE_TILE(acc1, 1)
    GCN_STORE_TILE(acc2, 2) GCN_STORE_TILE(acc3, 3)
    GCN_STORE_TILE(acc4, 4) GCN_STORE_TILE(acc5, 5)
    GCN_STORE_TILE(acc6, 6) GCN_STORE_TILE(acc7, 7)
#undef GCN_STORE_TILE
}

// ---------------------------------------------------------------------------
// Edge scatter: one wave per edge; lane handles 4 channels (float4 load,
// 4x native f32 atomic add into acc[dst]).  L2-resident working set.
// ---------------------------------------------------------------------------
__global__ void gcn_scatter_kernel(const int* __restrict__ src,
                                   const int* __restrict__ dst,
                                   const float* __restrict__ dinv,
                                   const float* __restrict__ xw,
                                   float* __restrict__ acc, int E) {
    int e = blockIdx.x * (blockDim.x >> 5) + (threadIdx.x >> 5);
    if (e >= E) return;
    const int lane = threadIdx.x & 31;
    const int s = src[e];
    const int d = dst[e];
    const float norm = dinv[s] * dinv[d];
    const float4 v = ((const float4*)(xw + s * CHN))[lane];
    float* ad = acc + d * CHN + lane * 4;
    atom_add_f32(ad + 0, v.x * norm);
    atom_add_f32(ad + 1, v.y * norm);
    atom_add_f32(ad + 2, v.z * norm);
    atom_add_f32(ad + 3, v.w * norm);
}

// fuse: self-loop contribution + bias + ReLU, in place on acc
__global__ void gcn_finalize_kernel(float* __restrict__ acc,
                                    const float* __restrict__ xw,
                                    const float* __restrict__ dinv,
                                    const float* __restrict__ bias, int total) {
    int idx = blockIdx.x * blockDim.x + threadIdx.x;
    if (idx >= total) return;
    const int i = idx >> 7;         // node
    const int c = idx & (CHN - 1);  // channel
    const float di = dinv[i];
    const float v = acc[idx] + xw[idx] * di * di + bias[c];
    acc[idx] = fmaxf(v, 0.0f);
}

// ---------------------------------------------------------------------------
// Graph count: one thread per node
// ---------------------------------------------------------------------------
__global__ void gcn_cnt_kernel(const int* __restrict__ batch, unsigned* __restrict__ cnt, int n) {
    int i = blockIdx.x * blockDim.x + threadIdx.x;
    if (i < n) atomicAdd(&cnt[batch[i]], 1u);
}

// Pooling: 128 threads (one per channel) sweep a run of nodes; batch is
// sorted, so keep a running sum per graph-id and flush on change.
__global__ void gcn_pool_kernel(const float* __restrict__ h,
                                const int* __restrict__ batch,
                                float* __restrict__ pool, int n, int nodesPerBlock) {
    const int c  = threadIdx.x;          // 0..127
    const int n0 = blockIdx.x * nodesPerBlock;
    if (n0 >= n) return;
    const int n1 = min(n0 + nodesPerBlock, n);
    int cur = batch[n0];
    float s = 0.0f;
    for (int nd = n0; nd < n1; ++nd) {
        const int g = batch[nd];
        if (g != cur) {
            atom_add_f32(&pool[cur * CHN + c], s);
            s = 0.0f;
            cur = g;
        }
        s += h[nd * CHN + c];
    }
    atom_add_f32(&pool[cur * CHN + c], s);
}

// Final tiny linear: out[g][o] = mean_pool[g] . Wlin[:,o] + blin[o]
__global__ void gcn_head_kernel(const float* __restrict__ pool,
                                const unsigned* __restrict__ cnt,
                                const float* __restrict__ Wlin,
                                const float* __restrict__ blin,
                                float* __restrict__ out) {
    const int t = threadIdx.x;
    if (t >= NUM_G * OUT_C) return;
    const int g = t / OUT_C;
    const int o = t % OUT_C;
    const float inv = 1.0f / fmaxf((float)cnt[g], 1.0f);
    float s = blin[o];
    for (int c = 0; c < CHN; ++c)
        s += pool[g * CHN + c] * inv * Wlin[c * OUT_C + o];
    out[g * OUT_C + o] = s;
}

// ---------------------------------------------------------------------------
extern "C" void kernel_launch(void* const* d_in, const int* in_sizes, int n_in,
                              void* d_out, int out_size, void* d_ws, size_t ws_size,
                              hipStream_t stream) {
    const float* x    = (const float*)d_in[0];
    const int*   ei   = (const int*)d_in[1];       // [2, E] int32
    const int*   bat  = (const int*)d_in[2];       // [N]
    const float* W1   = (const float*)d_in[3];
    const float* b1   = (const float*)d_in[4];
    const float* W2   = (const float*)d_in[5];
    const float* b2   = (const float*)d_in[6];
    const float* Wlin = (const float*)d_in[7];
    const float* blin = (const float*)d_in[8];
    float* out = (float*)d_out;

    const int N = N_NODES, E = N_EDGES;
    const int* esrc = ei;
    const int* edst = ei + E;

    // workspace layout
    char* ws = (char*)d_ws;
    float*    bufA = (float*)ws;                               // N*128 f32 (xw)
    float*    bufB = (float*)(ws + (size_t)N * CHN * 4);       // N*128 f32 (acc/h)
    float*    dinv = (float*)(ws + (size_t)2 * N * CHN * 4);   // N f32
    unsigned* deg  = (unsigned*)(ws + (size_t)2 * N * CHN * 4 + (size_t)N * 4);
    float*    pool = (float*)((char*)deg + (size_t)N * 4);     // 64*128 f32
    unsigned* cnt  = (unsigned*)((char*)pool + NUM_G * CHN * 4);

    const size_t featBytes = (size_t)N * CHN * sizeof(float);

    // 1) zero deg / pool / cnt
    hipMemsetAsync(deg,  0, (size_t)N * 4, stream);
    hipMemsetAsync(pool, 0, NUM_G * CHN * 4, stream);
    hipMemsetAsync(cnt,  0, NUM_G * 4, stream);

    // 2) degree + dinv
    gcn_deg_kernel<<<(E + 255) / 256, 256, 0, stream>>>(edst, deg, E);
    gcn_dinv_kernel<<<(N + 255) / 256, 256, 0, stream>>>(deg, dinv, N);

    // ---- layer 1 ----
    {
        int strips = N / 16;                        // 6250
        int blocks = (strips + 7) / 8;
        gcn_gemm128_wmma_kernel<<<blocks, 256, 0, stream>>>(x, W1, bufA, N);
    }
    hipMemsetAsync(bufB, 0, featBytes, stream);
    gcn_scatter_kernel<<<(E + 7) / 8, 256, 0, stream>>>(esrc, edst, dinv, bufA, bufB, E);
    gcn_finalize_kernel<<<(N * CHN + 255) / 256, 256, 0, stream>>>(bufB, bufA, dinv, b1, N * CHN);

    // ---- layer 2 ----
    {
        int strips = N / 16;
        int blocks = (strips + 7) / 8;
        gcn_gemm128_wmma_kernel<<<blocks, 256, 0, stream>>>(bufB, W2, bufA, N);
    }
    hipMemsetAsync(bufB, 0, featBytes, stream);
    gcn_scatter_kernel<<<(E + 7) / 8, 256, 0, stream>>>(esrc, edst, dinv, bufA, bufB, E);
    gcn_finalize_kernel<<<(N * CHN + 255) / 256, 256, 0, stream>>>(bufB, bufA, dinv, b2, N * CHN);

    // ---- pooling + head ----
    gcn_cnt_kernel<<<(N + 255) / 256, 256, 0, stream>>>(bat, cnt, N);
    {
        const int nodesPerBlock = 256;
        int blocks = (N + nodesPerBlock - 1) / nodesPerBlock;
        gcn_pool_kernel<<<blocks, CHN, 0, stream>>>(bufB, bat, pool, N, nodesPerBlock);
    }
    gcn_head_kernel<<<1, NUM_G * OUT_C, 0, stream>>>(pool, cnt, Wlin, blin, out);
}